// CapsuleLayer2_30545807409362
// MI455X (gfx1250) — compile-verified
//
#include <hip/hip_runtime.h>
#include <hip/hip_bf16.h>

// CapsuleLayer dynamic routing, fused per-j recompute formulation for MI455X.
// B=32 batch, J=2048 input caps, I=16 in-dim, N=64 out caps, D=32 out-dim.
//
// u_hat[b,n,j,d] = sum_i W[n,j,d,i] * x[b,j,i]  -- never materialized in HBM.
// Each workgroup owns one j; W[:,j,:,:] is read once from HBM per routing pass
// (re-read of a 16KB chunk in sweep 2 hits L2). s[b,n,d] accumulated with
// global f32 atomics (s = 256KB, L2 resident). GEMM done with fp32 WMMA
// (v_wmma_f32_16x16x4_f32) for exact fp32 accumulation.

#define BB 32
#define JJ 2048
#define II 16
#define NN 64
#define DD 32
#define NCH 8                 // capsules n per chunk == waves per block
#define NCHUNKS (NN / NCH)    // 8
#define IPAD 18               // padded K stride (floats) to dodge LDS bank conflicts
#define THREADS 256

typedef float v2f __attribute__((ext_vector_type(2)));
typedef float v8f __attribute__((ext_vector_type(8)));

__device__ __forceinline__ float lane16_reduce(float x) {
  // sum within each 16-lane half of the wave32
  x += __shfl_xor(x, 1, 32);
  x += __shfl_xor(x, 2, 32);
  x += __shfl_xor(x, 4, 32);
  x += __shfl_xor(x, 8, 32);
  return x;
}

// Cooperative load of W[n0..n0+7, j, 0..31, 0..15] into padded LDS.
// 4096 floats, 16 per thread, b128 global loads, 8B LDS stores.
__device__ __forceinline__ void load_W_chunk(const float* __restrict__ W,
                                             float* __restrict__ Ws,
                                             int j, int c, int tid) {
  const int f0  = tid * 16;          // 256 threads * 16 floats = 4096
  const int nl  = f0 >> 9;           // 512 floats per n
  const int rem = f0 & 511;          // multiple of 16 -> i==0
  const int d   = rem >> 4;
  const float* gsrc = W + ((size_t)(c * NCH + nl) * JJ + j) * (DD * II) + rem;
  float* ldst = &Ws[(nl * DD + d) * IPAD];
#pragma unroll
  for (int q = 0; q < 4; ++q) {
    float4 t4 = *(const float4*)(gsrc + q * 4);
    v2f lo = {t4.x, t4.y};
    v2f hi = {t4.z, t4.w};
    *(v2f*)(ldst + q * 4)     = lo;
    *(v2f*)(ldst + q * 4 + 2) = hi;
  }
}

// Per-wave GEMM for its capsule n: acc[mt][dt] = x(16x16 tile mt) @ W(16-col tile dt)
// using four chained K=4 fp32 WMMAs per tile (K=16 total).
__device__ __forceinline__ void gemm_chunk(const float* __restrict__ xs,
                                           const float* __restrict__ Ws,
                                           int wave, int lane, v8f acc[2][2]) {
  const int hf = lane >> 4;    // 0: K lanes 0..1, 1: K lanes 2..3 (fp32 WMMA layout)
  const int lr = lane & 15;
  v8f z = {0.f, 0.f, 0.f, 0.f, 0.f, 0.f, 0.f, 0.f};
  acc[0][0] = z; acc[0][1] = z; acc[1][0] = z; acc[1][1] = z;
#pragma unroll
  for (int k0 = 0; k0 < II; k0 += 4) {
    const int ko = k0 + 2 * hf;
    v2f a0 = *(const v2f*)&xs[(lr) * IPAD + ko];          // A rows M=0..15 (batch)
    v2f a1 = *(const v2f*)&xs[(16 + lr) * IPAD + ko];     // A rows M=16..31
    v2f b0 = *(const v2f*)&Ws[(wave * DD + lr) * IPAD + ko];        // cols d=0..15
    v2f b1 = *(const v2f*)&Ws[(wave * DD + 16 + lr) * IPAD + ko];   // cols d=16..31
    acc[0][0] = __builtin_amdgcn_wmma_f32_16x16x4_f32(false, a0, false, b0,
                                                      (short)0, acc[0][0], false, false);
    acc[0][1] = __builtin_amdgcn_wmma_f32_16x16x4_f32(false, a0, false, b1,
                                                      (short)0, acc[0][1], false, false);
    acc[1][0] = __builtin_amdgcn_wmma_f32_16x16x4_f32(false, a1, false, b0,
                                                      (short)0, acc[1][0], false, false);
    acc[1][1] = __builtin_amdgcn_wmma_f32_16x16x4_f32(false, a1, false, b1,
                                                      (short)0, acc[1][1], false, false);
  }
}

// One routing pass. ITER==0: uniform coupling c=1/N, single sweep.
// ITER>=1: sweep1 computes logits (+ stored logits), softmax over N, sweep2
// recomputes u_hat and atomically accumulates s.
template <int ITER>
__global__ __launch_bounds__(THREADS) void caps_pass(
    const float* __restrict__ x, const float* __restrict__ W,
    const float* __restrict__ v_in, float* __restrict__ blog,
    float* __restrict__ s) {
  __shared__ __align__(16) float xs[BB * IPAD];          //  2.3 KB
  __shared__ __align__(16) float Ws[NCH * DD * IPAD];    // 18.0 KB
  __shared__ __align__(16) float vs[NCH * BB * DD];      // 32.0 KB (ITER>=1)
  __shared__ __align__(16) float lg[BB * 65];            //  8.3 KB (padded rows)

  const int tid  = threadIdx.x;
  const int lane = tid & 31;
  const int wave = tid >> 5;       // 0..7 -> owns capsule n = c*8 + wave
  const int j    = blockIdx.x;     // 0..2047
  const int hf   = lane >> 4;
  const int lr   = lane & 15;

  // x[:, j, :] -> LDS (512 floats)
  for (int idx = tid; idx < BB * II; idx += THREADS) {
    int b = idx >> 4, i = idx & 15;
    xs[b * IPAD + i] = x[((size_t)b * JJ + j) * II + i];
  }

  // ---------------- Sweep 1: logits + softmax (ITER >= 1) ----------------
  if (ITER >= 1) {
    for (int c = 0; c < NCHUNKS; ++c) {
      __syncthreads();
      load_W_chunk(W, Ws, j, c, tid);
      // v chunk: vs[(nl*32 + b)*32 + d] = v_in[b, c*8+nl, d]
      for (int idx = tid; idx < NCH * BB * DD; idx += THREADS) {
        int nl = idx >> 10, b = (idx >> 5) & 31, d = idx & 31;
        vs[idx] = v_in[((size_t)b * NN + c * NCH + nl) * DD + d];
      }
      if (c + 1 < NCHUNKS) {  // prefetch next W chunk (global_prefetch_b8)
        int f0 = tid * 16, nl = f0 >> 9, rem = f0 & 511;
        __builtin_prefetch(
            W + ((size_t)((c + 1) * NCH + nl) * JJ + j) * (DD * II) + rem, 0, 1);
      }
      __syncthreads();

      v8f acc[2][2];
      gemm_chunk(xs, Ws, wave, lane, acc);
      const int n = c * NCH + wave;
#pragma unroll
      for (int mt = 0; mt < 2; ++mt) {
#pragma unroll
        for (int r = 0; r < 8; ++r) {
          const int b = mt * 16 + r + 8 * hf;  // batch held by this lane half
          float p = acc[mt][0][r] * vs[(wave * BB + b) * DD + lr] +
                    acc[mt][1][r] * vs[(wave * BB + b) * DD + 16 + lr];
          p = lane16_reduce(p);                // sum over d within each half
          if (lr == 0) lg[b * 65 + n] = p;     // raw logit for (b, n, j)
        }
      }
    }
    __syncthreads();

    if (ITER == 2) {  // add logits accumulated in the previous iteration
      for (int idx = tid; idx < BB * NN; idx += THREADS) {
        int b = idx >> 6, n = idx & 63;
        lg[b * 65 + n] += blog[((size_t)b * JJ + j) * NN + n];
      }
      __syncthreads();
    }
    if (ITER == 1) {  // persist b1 = v0 . u_hat for the next pass
      for (int idx = tid; idx < BB * NN; idx += THREADS) {
        int b = idx >> 6, n = idx & 63;
        blog[((size_t)b * JJ + j) * NN + n] = lg[b * 65 + n];
      }
      __syncthreads();
    }

    // softmax over n (64) per batch row; result c stored back into lg
    if (tid < BB) {
      const int b = tid;
      float m = -1e30f;
      for (int n = 0; n < NN; ++n) m = fmaxf(m, lg[b * 65 + n]);
      float ssum = 0.f;
      for (int n = 0; n < NN; ++n) {
        float e = __expf(lg[b * 65 + n] - m);
        lg[b * 65 + n] = e;
        ssum += e;
      }
      float inv = 1.f / ssum;
      for (int n = 0; n < NN; ++n) lg[b * 65 + n] *= inv;
    }
    __syncthreads();
  }

  // ------------- Sweep 2: s[b,n,d] += c * u_hat (atomic f32) -------------
  for (int c = 0; c < NCHUNKS; ++c) {
    __syncthreads();
    load_W_chunk(W, Ws, j, c, tid);  // L2 hit on ITER>=1 second read
    if (c + 1 < NCHUNKS) {
      int f0 = tid * 16, nl = f0 >> 9, rem = f0 & 511;
      __builtin_prefetch(
          W + ((size_t)((c + 1) * NCH + nl) * JJ + j) * (DD * II) + rem, 0, 1);
    }
    __syncthreads();

    v8f acc[2][2];
    gemm_chunk(xs, Ws, wave, lane, acc);
    const int n = c * NCH + wave;
#pragma unroll
    for (int mt = 0; mt < 2; ++mt) {
#pragma unroll
      for (int r = 0; r < 8; ++r) {
        const int b = mt * 16 + r + 8 * hf;
        const float cv = (ITER == 0) ? (1.f / (float)NN) : lg[b * 65 + n];
#pragma unroll
        for (int dt = 0; dt < 2; ++dt) {
          const int d = dt * 16 + lr;
          atomicAdd(&s[((size_t)b * NN + n) * DD + d], cv * acc[mt][dt][r]);
        }
      }
    }
  }
}

__global__ __launch_bounds__(THREADS) void caps_squash(
    const float* __restrict__ s, float* __restrict__ v) {
  int idx = blockIdx.x * blockDim.x + threadIdx.x;  // (b*N + n)
  if (idx >= BB * NN) return;
  const float* row = s + (size_t)idx * DD;
  float s2 = 0.f;
#pragma unroll
  for (int d = 0; d < DD; ++d) { float t = row[d]; s2 += t * t; }
  float scale = s2 / ((1.f + s2) * sqrtf(s2 + 1e-7f));
  float* out = v + (size_t)idx * DD;
#pragma unroll
  for (int d = 0; d < DD; ++d) out[d] = row[d] * scale;
}

__global__ __launch_bounds__(THREADS) void caps_zero(float* __restrict__ p, int n) {
  int i = blockIdx.x * blockDim.x + threadIdx.x;
  if (i < n) p[i] = 0.f;
}

extern "C" void kernel_launch(void* const* d_in, const int* in_sizes, int n_in,
                              void* d_out, int out_size, void* d_ws, size_t ws_size,
                              hipStream_t stream) {
  (void)in_sizes; (void)n_in; (void)out_size; (void)ws_size;
  const float* x = (const float*)d_in[0];   // [32, 2048, 16]
  const float* W = (const float*)d_in[1];   // [64, 2048, 32, 16]
  float* out = (float*)d_out;               // [32, 64, 32]

  // workspace layout (floats): s | v | blog
  float* s    = (float*)d_ws;               // 65536
  float* v    = s + BB * NN * DD;           // 65536
  float* blog = v + BB * NN * DD;           // 32*2048*64 = 4,194,304

  const int svN = BB * NN * DD;
  const dim3 zg((svN + THREADS - 1) / THREADS), zb(THREADS);
  const dim3 pg(JJ), pb(THREADS);
  const dim3 qg((BB * NN + THREADS - 1) / THREADS), qb(THREADS);

  // routing iteration 0: uniform coupling
  caps_zero<<<zg, zb, 0, stream>>>(s, svN);
  caps_pass<0><<<pg, pb, 0, stream>>>(x, W, nullptr, blog, s);
  caps_squash<<<qg, qb, 0, stream>>>(s, v);

  // routing iteration 1
  caps_zero<<<zg, zb, 0, stream>>>(s, svN);
  caps_pass<1><<<pg, pb, 0, stream>>>(x, W, v, blog, s);
  caps_squash<<<qg, qb, 0, stream>>>(s, v);

  // routing iteration 2 -> output
  caps_zero<<<zg, zb, 0, stream>>>(s, svN);
  caps_pass<2><<<pg, pb, 0, stream>>>(x, W, v, blog, s);
  caps_squash<<<qg, qb, 0, stream>>>(s, out);
}